// ParabolicPool1DFast_27419071218093
// MI455X (gfx1250) — compile-verified
//
#include <hip/hip_runtime.h>
#include <cstdint>
#include <cstddef>

// ParabolicPool1D (max-plus dilation, ks=7, stride=2), f32.
// Memory-bound: 402 MB traffic -> ~17us floor at 23.3 TB/s. Compute is ~1000x
// below roofline, so the kernel is pure data movement:
//   TDM tensor_load_to_lds (TENSORcnt) streams each 16.4KB row tile into LDS
//   (fallback: per-lane global_load_async_to_lds_b128 on ASYNCcnt),
//   then 8 wave32s compute 8 outputs/lane and store nontemporal float4.

#define L_IN      32768
#define L_OUT     16384
#define NCH       256
#define TILE_OUT  2048
#define TILE_IN   (2 * TILE_OUT + 8)     // 4104 floats; tile[0] <-> g0 = 2*o0 - 4
#define NGROUPS   (TILE_IN / 4)          // 1026 b128 groups (fallback path)
#define BLOCK     256                    // 8 wave32s

typedef __attribute__((ext_vector_type(4))) float        v4f;
typedef __attribute__((ext_vector_type(4))) unsigned int u32x4;
typedef __attribute__((ext_vector_type(8))) int          i32x8;
typedef __attribute__((ext_vector_type(4))) int          i32x4;

#define HAVE_TDM __has_builtin(__builtin_amdgcn_tensor_load_to_lds)

__device__ __forceinline__ unsigned int sget(unsigned int x) {
    return __builtin_amdgcn_readfirstlane(x);
}

__global__ __launch_bounds__(BLOCK)
void parapool1d_kernel(const float* __restrict__ f,
                       const float* __restrict__ t,
                       float* __restrict__ out)
{
    __shared__ __attribute__((aligned(16))) float tile[TILE_IN];

    const int tid = threadIdx.x;
    const int row = blockIdx.y;                 // row = b*NCH + c
    const int c   = row & (NCH - 1);
    const int o0  = blockIdx.x * TILE_OUT;
    const int g0  = 2 * o0 - 4;                 // global float idx of tile[0]

    const float* frow = f   + (size_t)row * L_IN;
    float*       orow = out + (size_t)row * L_OUT;

    // Low 32 bits of the generic (flat) address of static LDS == LDS byte offset.
    const uint32_t lds_base = (uint32_t)(uintptr_t)&tile[0];

#if HAVE_TDM
    // ---- Tensor Data Mover: one 1D DMA per tile, issued by wave 0 only.
    if (tid < 32) {
        const int gstart = g0 < 0 ? 0 : g0;               // trim left halo (16B)
        const int skip   = gstart - g0;                    // 0 or 4 elements
        const uint32_t ndst   = (uint32_t)(TILE_IN - skip);   // tile_dim0
        const uint32_t remain = (uint32_t)(L_IN - gstart);    // tensor_dim0 (OOB->0)
        const uint64_t ga     = (uint64_t)(uintptr_t)(frow + gstart);
        const uint32_t ldsa   = lds_base + (uint32_t)(skip * 4);

        // D# group 0: count=1 | lds_addr | global_addr[56:0] | type=2
        u32x4 d0;
        d0.x = sget(1u);
        d0.y = sget(ldsa);
        d0.z = sget((uint32_t)(ga & 0xFFFFFFFFu));
        d0.w = sget(((uint32_t)(ga >> 32) & 0x01FFFFFFu) | 0x80000000u);

        // D# group 1: data_size=2 (4B); tensor_dim0=remain; tensor_dim1=1;
        // tile_dim0=ndst; tile_dim1/2=0 (unused); dim0_stride=remain.
        i32x8 d1;
        d1[0] = (int)sget(2u << 16);                              // data_size=4B
        d1[1] = (int)sget((remain & 0xFFFFu) << 16);              // td0[15:0]
        d1[2] = (int)sget(((remain >> 16) & 0xFFFFu) | (1u << 16)); // td0[31:16], td1=1
        d1[3] = (int)sget(ndst << 16);                            // tile_dim0
        d1[4] = 0;                                                // tile_dim1/2 unused
        d1[5] = (int)sget(remain);                                // dim0_stride[31:0]
        d1[6] = 0;                                                // stride hi / s1 lo
        d1[7] = 0;

        i32x4 dz4; dz4[0] = 0; dz4[1] = 0; dz4[2] = 0; dz4[3] = 0;   // groups 2/3: 1D
        i32x8 dz8;
        dz8[0] = 0; dz8[1] = 0; dz8[2] = 0; dz8[3] = 0;
        dz8[4] = 0; dz8[5] = 0; dz8[6] = 0; dz8[7] = 0;

        __builtin_amdgcn_tensor_load_to_lds(d0, d1, dz4, dz4, dz8, 0);
        __builtin_amdgcn_s_wait_tensorcnt((short)0);
    }
#else
    // ---- Fallback: per-lane async copy global -> LDS, 16B per op, clamped.
    for (int k = tid; k < NGROUPS; k += BLOCK) {
        int s = g0 + 4 * k;
        s = s < 0 ? 0 : (s > (L_IN - 4) ? (L_IN - 4) : s);
        const uint32_t ldst = lds_base + (uint32_t)(k * 16);
        const int      boff = s * 4;
        asm volatile("global_load_async_to_lds_b128 %0, %1, %2"
                     :
                     : "v"(ldst), "v"(boff), "s"(frow)
                     : "memory");
    }
    asm volatile("s_wait_asynccnt 0" ::: "memory");
#endif

    // ---- -inf halo fixup (OOB-left never loaded; OOB-right zero-filled by TDM)
    if (tid < 8) {
        const int idx = (tid < 4) ? tid : (TILE_IN - 8 + tid);  // 0..3, 4100..4103
        const int g = g0 + idx;
        if (g < 0 || g >= L_IN) tile[idx] = -__builtin_inff();
    }
    __syncthreads();

    // ---- per-channel parabolic taps: h[j] = -(j-3)^2 / (4*t[c]), h[3] = 0
    const float tc  = t[c];
    const float inv = 1.0f / (4.0f * tc);
    const float h1  = -1.0f * inv;   // |z| = 1
    const float h2  = -4.0f * inv;   // |z| = 2
    const float h3  = -9.0f * inv;   // |z| = 3

    // out[o] = max_j tile[2*lo + j + 1] + h[j], lo = o - o0
    #pragma unroll
    for (int half = 0; half < 2; ++half) {
        const int lo = tid * 4 + half * (BLOCK * 4);   // 4 consecutive outputs
        float w[13];
        #pragma unroll
        for (int i = 0; i < 13; ++i) w[i] = tile[2 * lo + 1 + i];

        float r[4];
        #pragma unroll
        for (int q = 0; q < 4; ++q) {
            const int b = 2 * q;
            float m = w[b + 3];                                   // center tap, h=0
            m = fmaxf(m, fmaxf(w[b + 2], w[b + 4]) + h1);
            m = fmaxf(m, fmaxf(w[b + 1], w[b + 5]) + h2);
            m = fmaxf(m, fmaxf(w[b + 0], w[b + 6]) + h3);
            r[q] = m;
        }
        v4f v; v.x = r[0]; v.y = r[1]; v.z = r[2]; v.w = r[3];
        __builtin_nontemporal_store(v, (v4f*)(orow + o0 + lo));
    }
}

extern "C" void kernel_launch(void* const* d_in, const int* in_sizes, int n_in,
                              void* d_out, int out_size, void* d_ws, size_t ws_size,
                              hipStream_t stream) {
    (void)n_in; (void)out_size; (void)d_ws; (void)ws_size;
    const float* f = (const float*)d_in[0];
    const float* t = (const float*)d_in[1];
    float* out     = (float*)d_out;

    const int rows = in_sizes[0] / L_IN;       // B*C = 2048
    dim3 grid(L_OUT / TILE_OUT, rows, 1);      // 8 x 2048 workgroups
    parapool1d_kernel<<<grid, BLOCK, 0, stream>>>(f, t, out);
}